// VectorQuantizer_1958505087081
// MI455X (gfx1250) — compile-verified
//
#include <hip/hip_runtime.h>

typedef __attribute__((ext_vector_type(2))) float v2f;
typedef __attribute__((ext_vector_type(4))) float v4f;
typedef __attribute__((ext_vector_type(8))) float v8f;

#define NUM_LATENTS 8192
#define LATENT_DIM  256
#define NTOK        (32 * 512)

#define TILE_N      16
#define NTILES      (NUM_LATENTS / TILE_N)     // 512
#define WAVES_PER_WG 4
#define THREADS     (WAVES_PER_WG * 32)        // 128
#define TOK_PER_WG  (WAVES_PER_WG * 16)        // 64
#define LDS_STRIDE  260                        // 256 + 4 pad -> conflict-free ds_load_b64
#define STAGE_V4    ((TILE_N * LATENT_DIM / 4) / THREADS)   // 8 float4 per thread

// ---------------------------------------------------------------------------
// w2[k] = sum_d codebook[k][d]^2   (one wave per codebook row)
// ---------------------------------------------------------------------------
__global__ __launch_bounds__(256) void vq_w2_kernel(const float* __restrict__ cb,
                                                    float* __restrict__ w2) {
    const int wave = (blockIdx.x * blockDim.x + threadIdx.x) >> 5;
    const int lane = threadIdx.x & 31;
    if (wave >= NUM_LATENTS) return;
    const float* row = cb + (size_t)wave * LATENT_DIM;
    float s = 0.0f;
#pragma unroll
    for (int i = 0; i < LATENT_DIM / 32; ++i) {
        float v = row[lane + i * 32];
        s = fmaf(v, v, s);
    }
#pragma unroll
    for (int m = 16; m >= 1; m >>= 1) s += __shfl_xor(s, m, 32);
    if (lane == 0) w2[wave] = s;
}

// ---------------------------------------------------------------------------
// Main kernel: per token, argmin_k( w2[k] - 2 x.w[k] ) via f32 WMMA with a
// double-buffered, software-pipelined LDS codebook tile and a depth-4
// rotating B-fragment register pipeline, then gather + write (z_q, z, x, idx).
//
// Wave layout (wave32, V_WMMA_F32_16X16X4_F32):
//   A (16x4, M=tokens): lane L<16 -> M=L, K={0,1}; lane L>=16 -> M=L-16,
//                       K={2,3}. All 64 K-steps of -2*x resident in 128 VGPRs.
//   B (4x16, N=codebook entries): lane L<16 -> N=L, K={0,1}; lane>=16 ->
//                       N=L-16, K={2,3}. Read from LDS tile.
//   C/D (16x16): VGPR s, lanes 0-15 -> (M=s, N=lane); lanes 16-31 -> (M=s+8).
//   C pre-loaded with w2[n] so D is the final squared-distance score.
// ---------------------------------------------------------------------------
__global__ __launch_bounds__(THREADS) void vq_main_kernel(const float* __restrict__ x,
                                                          const float* __restrict__ cb,
                                                          const float* __restrict__ w2,
                                                          float* __restrict__ out) {
    __shared__ __align__(16) float ldsB[2][TILE_N * LDS_STRIDE];
    __shared__ int idxbuf[TOK_PER_WG];

    const int tid  = threadIdx.x;
    const int lane = tid & 31;
    const int wv   = tid >> 5;
    const int nloc = lane & 15;   // token slot (A) / codebook column (B,D)
    const int half = lane >> 4;   // K half selector

    const int tok0 = blockIdx.x * TOK_PER_WG;
    const int T    = tok0 + wv * 16 + nloc;   // this lane's token row

    // ---- Preload A panel: -2 * x[T][*] swizzled into WMMA A fragments.
    v2f a[64];
    {
        const float* xrow = x + (size_t)T * LATENT_DIM + 2 * half;
#pragma unroll
        for (int kk = 0; kk < 64; ++kk) {
            v2f v = *(const v2f*)(xrow + 4 * kk);
            a[kk] = v * -2.0f;
        }
    }

    float minval[8];
    int   minidx[8];
#pragma unroll
    for (int s = 0; s < 8; ++s) { minval[s] = 3.4e38f; minidx[s] = 0; }

    // ---- Pipeline prolog: tile 0 -> regs -> LDS buf 0, plus its w2 value.
    v4f stage[STAGE_V4];
#pragma unroll
    for (int i = 0; i < STAGE_V4; ++i) {
        const int f   = tid + i * THREADS;   // float4 id within tile
        const int row = f >> 6;              // 64 float4 per row
        const int c4  = f & 63;
        stage[i] = *(const v4f*)(cb + (size_t)row * LATENT_DIM + 4 * c4);
    }
    float w2_cur = w2[nloc];
#pragma unroll
    for (int i = 0; i < STAGE_V4; ++i) {
        const int f   = tid + i * THREADS;
        const int row = f >> 6;
        const int c4  = f & 63;
        *(v4f*)(&ldsB[0][row * LDS_STRIDE + 4 * c4]) = stage[i];
    }
    __syncthreads();

    for (int tile = 0; tile < NTILES; ++tile) {
        const int base_n = tile * TILE_N;
        const bool have_next = (tile + 1 < NTILES);
        float w2_next = 0.0f;

        // Issue next tile's global loads up front; latency hides under WMMAs.
        if (have_next) {
            const float* nb = cb + (size_t)(base_n + TILE_N) * LATENT_DIM;
#pragma unroll
            for (int i = 0; i < STAGE_V4; ++i) {
                const int f   = tid + i * THREADS;
                const int row = f >> 6;
                const int c4  = f & 63;
                stage[i] = *(const v4f*)(nb + (size_t)row * LATENT_DIM + 4 * c4);
            }
            w2_next = w2[base_n + TILE_N + nloc];
        }

        // ---- 64-step WMMA chain, depth-4 rotating B-fragment pipeline.
        const float* bbase = &ldsB[tile & 1][nloc * LDS_STRIDE + 2 * half];
        v8f c;
#pragma unroll
        for (int s = 0; s < 8; ++s) c[s] = w2_cur;

        v2f bf[4];
#pragma unroll
        for (int i = 0; i < 4; ++i) bf[i] = *(const v2f*)(bbase + 4 * i);
#pragma unroll
        for (int kk = 0; kk < 64; ++kk) {
            const v2f bcur = bf[kk & 3];
            if (kk + 4 < 64) bf[kk & 3] = *(const v2f*)(bbase + 4 * (kk + 4));
            c = __builtin_amdgcn_wmma_f32_16x16x4_f32(
                    /*neg_a=*/false, a[kk], /*neg_b=*/false, bcur,
                    /*c_mod=*/(short)0, c, /*reuse_a=*/false, /*reuse_b=*/false);
        }

        // Running argmin (strict < keeps the lowest index, matching jnp.argmin).
        const int myn = base_n + nloc;
#pragma unroll
        for (int s = 0; s < 8; ++s) {
            const float v = c[s];
            if (v < minval[s]) { minval[s] = v; minidx[s] = myn; }
        }

        // ---- Stage next tile into the other buffer; one barrier per tile.
        if (have_next) {
#pragma unroll
            for (int i = 0; i < STAGE_V4; ++i) {
                const int f   = tid + i * THREADS;
                const int row = f >> 6;
                const int c4  = f & 63;
                *(v4f*)(&ldsB[(tile + 1) & 1][row * LDS_STRIDE + 4 * c4]) = stage[i];
            }
            w2_cur = w2_next;
            __syncthreads();
        }
    }

    // Cross-lane argmin over the 16 columns in each half-wave.
#pragma unroll
    for (int s = 0; s < 8; ++s) {
        float v  = minval[s];
        int   id = minidx[s];
#pragma unroll
        for (int m = 1; m <= 8; m <<= 1) {
            const float ov = __shfl_xor(v, m, 32);
            const int   oi = __shfl_xor(id, m, 32);
            if (ov < v || (ov == v && oi < id)) { v = ov; id = oi; }
        }
        // lanes 0-15 slot s -> token m=s; lanes 16-31 -> token m=s+8
        if (nloc == 0) idxbuf[wv * 16 + s + half * 8] = id;
    }
    __syncthreads();

    // Outputs: d_out = [ z_q | z | x | indices ] flat fp32.
    const size_t ND = (size_t)NTOK * LATENT_DIM;
    for (int i = tid; i < TOK_PER_WG * LATENT_DIM; i += THREADS) {
        const int    j  = i >> 8;       // local token
        const int    d  = i & 255;
        const size_t g  = (size_t)(tok0 + j) * LATENT_DIM + d;
        const float  xv = x[g];
        const float  zv = cb[(size_t)idxbuf[j] * LATENT_DIM + d];
        out[g]          = xv + (zv - xv);  // z_q (straight-through, same fp order)
        out[ND + g]     = zv;              // z
        out[2 * ND + g] = xv;              // x passthrough
    }
    if (tid < TOK_PER_WG) {
        out[3 * ND + tok0 + tid] = (float)idxbuf[tid];  // indices
    }
}

// ---------------------------------------------------------------------------
extern "C" void kernel_launch(void* const* d_in, const int* in_sizes, int n_in,
                              void* d_out, int out_size, void* d_ws, size_t ws_size,
                              hipStream_t stream) {
    const float* x  = (const float*)d_in[0];   // [32,512,256] f32
    const float* cb = (const float*)d_in[1];   // [8192,256]   f32
    float* out = (float*)d_out;
    float* w2  = (float*)d_ws;                 // 8192 f32 scratch

    // One wave per codebook row: 8192 waves = 1024 blocks of 256 threads.
    vq_w2_kernel<<<(NUM_LATENTS * 32) / 256, 256, 0, stream>>>(cb, w2);

    // 64 tokens per workgroup, double-buffered codebook tiles.
    vq_main_kernel<<<NTOK / TOK_PER_WG, THREADS, 0, stream>>>(x, cb, w2, out);
}